// LocalFeatureEncoder_1580547966482
// MI455X (gfx1250) — compile-verified
//
#include <hip/hip_runtime.h>
#include <hip/hip_bf16.h>

// Problem constants (from reference): B=8, T=32768, K=24, Z=128, F=64
#define BB   8
#define TT   32768
#define KK   24
#define ZZ   128
#define FF   64

typedef __attribute__((ext_vector_type(2))) float v2f;
typedef __attribute__((ext_vector_type(8))) float v8f;

// ---------------------------------------------------------------------------
// Kernel 1: local[b,k,f] = sum_z W[k,f,z] * g[b,z] + bias[k,f]
// where g[b,:] = concat(shape(64), structure(32), pose(32)).
// Stored TRANSPOSED as localT[b][f][k] (f-major, k contiguous) so kernel 2's
// WMMA B-operand loads (k, k+1 per lane) are single aligned b64 loads.
// Tiny: 12288 dots of length 128 -> one 64-thread block per (b,k).
// ---------------------------------------------------------------------------
__global__ void local_encoder_kernel(const float* __restrict__ shape_code,
                                     const float* __restrict__ structure_code,
                                     const float* __restrict__ pose_code,
                                     const float* __restrict__ W,
                                     const float* __restrict__ bias,
                                     float* __restrict__ localT)
{
    const int bk = blockIdx.x;          // 0..B*K-1
    const int b  = bk / KK;
    const int k  = bk % KK;
    const int f  = threadIdx.x;         // 0..63

    __shared__ float g[ZZ];
    g[f] = shape_code[b * 64 + f];                      // z = 0..63
    if (f < 32) {
        g[64 + f] = structure_code[b * 32 + f];         // z = 64..95
        g[96 + f] = pose_code[b * 32 + f];              // z = 96..127
    }
    __syncthreads();

    const float* w = W + ((size_t)k * FF + f) * ZZ;
    float acc = bias[k * FF + f];
#pragma unroll 8
    for (int z = 0; z < ZZ; ++z)
        acc = fmaf(w[z], g[z], acc);

    localT[((size_t)b * FF + f) * KK + k] = acc;
}

// ---------------------------------------------------------------------------
// Kernel 2: out[b, t, f] = sum_k lbs[b,t,k] * localT[b][f][k]
// Per batch: (T x 24) x (24 x 64) GEMM on V_WMMA_F32_16X16X4_F32.
// K=24 == exactly 6 WMMA K-steps of 4, no padding.
// ---------------------------------------------------------------------------
#define CHUNKS        64                       // gridDim.y
#define MTILES        (TT / 16)                // 2048 M-tiles per batch
#define TILES_PER_BLK (MTILES / CHUNKS)        // 32
#define WAVES_PER_BLK 8

__global__ void lbs_wmma_kernel(const float* __restrict__ lbs,     // (B, T, 24)
                                const float* __restrict__ localT,  // (B, 64, 24)
                                float* __restrict__ out)           // (B, T, 64)
{
    const int b     = blockIdx.x;
    const int lane  = threadIdx.x & 31;
    const int wave  = threadIdx.x >> 5;
    const int mlo   = lane & 15;        // row/col-within-tile index
    const int khalf = lane >> 4;        // 0 or 1 -> k offset 2*khalf

    // --- Hoist B operand (localT for this batch) into registers: 4 N-tiles
    // x 6 K-steps. B-matrix layout (4x16 KxN, 2 VGPRs): lane l, vgpr j holds
    // (k_local = 2*(l>>4)+j, n = l&15)  ->  k, k+1 contiguous => b64 load.
    const float* lt = localT + (size_t)b * FF * KK;
    v2f Breg[4][6];
#pragma unroll
    for (int nt = 0; nt < 4; ++nt) {
#pragma unroll
        for (int s = 0; s < 6; ++s) {
            const int f = nt * 16 + mlo;
            const int k = s * 4 + 2 * khalf;
            Breg[nt][s] = *(const v2f*)(lt + (size_t)f * KK + k);   // 8B aligned
        }
    }

    const int tile0 = blockIdx.y * TILES_PER_BLK;

    for (int mt = tile0 + wave; mt < tile0 + TILES_PER_BLK; mt += WAVES_PER_BLK) {
        const int t0 = mt * 16;

        // A-matrix layout (16x4 MxK, 2 VGPRs): lane l, vgpr j holds
        // (m = l&15, k = 4s + 2*(l>>4) + j) -> k,k+1 contiguous => b64 load.
        const float* arow = lbs + ((size_t)b * TT + t0 + mlo) * KK + 2 * khalf;

        v8f acc0 = {}, acc1 = {}, acc2 = {}, acc3 = {};
#pragma unroll
        for (int s = 0; s < 6; ++s) {
            const v2f A = *(const v2f*)(arow + s * 4);               // 8B aligned
            acc0 = __builtin_amdgcn_wmma_f32_16x16x4_f32(false, A, false, Breg[0][s],
                                                         (short)0, acc0, false, false);
            acc1 = __builtin_amdgcn_wmma_f32_16x16x4_f32(false, A, false, Breg[1][s],
                                                         (short)0, acc1, false, false);
            acc2 = __builtin_amdgcn_wmma_f32_16x16x4_f32(false, A, false, Breg[2][s],
                                                         (short)0, acc2, false, false);
            acc3 = __builtin_amdgcn_wmma_f32_16x16x4_f32(false, A, false, Breg[3][s],
                                                         (short)0, acc3, false, false);
        }

        // C/D layout (16x16 f32, 8 VGPRs): vgpr v, lane l = (m = v + 8*(l>>4),
        // n = l&15). Each half-wave store is 64B contiguous.
        float* orow = out + ((size_t)b * TT + t0) * FF;
#pragma unroll
        for (int v = 0; v < 8; ++v) {
            const size_t m = (size_t)(v + 8 * khalf);
            orow[m * FF +  0 + mlo] = acc0[v];
            orow[m * FF + 16 + mlo] = acc1[v];
            orow[m * FF + 32 + mlo] = acc2[v];
            orow[m * FF + 48 + mlo] = acc3[v];
        }
    }
}

// ---------------------------------------------------------------------------
extern "C" void kernel_launch(void* const* d_in, const int* in_sizes, int n_in,
                              void* d_out, int out_size, void* d_ws, size_t ws_size,
                              hipStream_t stream) {
    (void)in_sizes; (void)n_in; (void)out_size; (void)ws_size;

    const float* shape_code     = (const float*)d_in[0];  // (B, 64)
    const float* structure_code = (const float*)d_in[1];  // (B, 32)
    const float* pose_code      = (const float*)d_in[2];  // (B, 32)
    const float* lbs_weights    = (const float*)d_in[3];  // (B, T, K)
    const float* W              = (const float*)d_in[4];  // (K, F, Z)
    const float* bias           = (const float*)d_in[5];  // (K, F)
    float*       out            = (float*)d_out;          // (B, T, F)
    float*       localT         = (float*)d_ws;           // (B, F, K) = 48 KB

    // Stage 1: per-joint linear -> localT (transposed for WMMA B-operand)
    local_encoder_kernel<<<dim3(BB * KK), dim3(FF), 0, stream>>>(
        shape_code, structure_code, pose_code, W, bias, localT);

    // Stage 2: bandwidth-bound weighted combine on the fp32 WMMA pipe
    lbs_wmma_kernel<<<dim3(BB, CHUNKS), dim3(32 * WAVES_PER_BLK), 0, stream>>>(
        lbs_weights, localT, out);
}